// BoneHypergraphModule_59261958750994
// MI455X (gfx1250) — compile-verified
//
#include <hip/hip_runtime.h>

// MI455X / gfx1250 fused hypergraph kernel.
// Key identity: Xr = H^T (H xf)  ==>  Xr[c,t,:] = P @ xf[c,t,:],  P = H^T H (21x21, symmetric, per-sample).
// So xf (352 MB) never touches HBM; it lives in LDS per (n, t-tile) block.
// fp32 end-to-end using V_WMMA_F32_16X16X4_F32 (exact-precision CDNA5 matrix path).
// Bulk x-tile staging uses GLOBAL_LOAD_ASYNC_TO_LDS_B32 (ASYNCcnt) — no VGPR round-trip.

#define EPS 1e-5f

typedef __attribute__((ext_vector_type(2))) float v2f;
typedef __attribute__((ext_vector_type(8))) float v8f;

#define N_   128
#define C_   128
#define T_   256
#define V_   21
#define E_   8
#define M_   13          // E + 5 finger masks
#define TT   16          // time steps per block in main kernel (16*21 = 336 = 21 col-tiles of 16)
#define NCOL (TT * V_)   // 336
#define VPAD 24          // V padded to multiple of 4 for stage-2 K dim
#define PCOLS 32         // V padded to 2 n-tiles of 16 for stage-2 N dim

// workspace layout (floats): xm then padded P
#define WS_XM 0
#define WS_P  (N_ * C_ * V_)

// async copy of one dword: global (64-bit addr) -> LDS (byte offset = low 32 bits of
// the generic pointer to a __shared__ object, per the gfx1250 LDS aperture mapping)
__device__ __forceinline__ void async_g2l_b32(const float* gaddr, const float* lds) {
  const unsigned lds_b = (unsigned)(unsigned long long)lds;
  asm volatile("global_load_async_to_lds_b32 %0, %1, off"
               :: "v"(lds_b), "v"((unsigned long long)gaddr)
               : "memory");
}

// ---------------------------------------------------------------- kernel 1
// xm[n,c,v] = mean over t of x[n,c,t,v].  One block per (n,c); thread t owns time t.
__global__ __launch_bounds__(256) void kmean(const float* __restrict__ x,
                                             float* __restrict__ xm) {
  __shared__ float s[V_];
  const int nc = blockIdx.x;         // n*C + c
  const int t  = threadIdx.x;        // 256 == T_
  if (t < V_) s[t] = 0.f;
  __syncthreads();
  const float* p = x + ((size_t)nc * T_ + t) * V_;
  float loc[V_];
#pragma unroll
  for (int v = 0; v < V_; ++v) loc[v] = p[v];
#pragma unroll
  for (int v = 0; v < V_; ++v) atomicAdd(&s[v], loc[v]);   // ds_add_f32
  __syncthreads();
  if (t < V_) xm[(size_t)nc * V_ + t] = s[t] * (1.0f / T_);
}

// ---------------------------------------------------------------- kernel 2
// Per sample: h = BN(dyn_w @ xm)+ReLU, softmax over V, concat finger masks -> H (13x21),
// then P = H^T H written zero-padded as [VPAD(K rows)][PCOLS(N cols)].
__global__ __launch_bounds__(256) void kedges(
    const float* __restrict__ xm,
    const float* __restrict__ dyn_w, const float* __restrict__ dyn_b,
    const float* __restrict__ dyn_gamma, const float* __restrict__ dyn_beta,
    const float* __restrict__ dyn_mean, const float* __restrict__ dyn_var,
    float* __restrict__ Ppad) {
  __shared__ float xs[C_ * V_];   // 2688 floats
  __shared__ float Hs[M_][V_];
  const int n = blockIdx.x, tid = threadIdx.x;

  for (int i = tid; i < C_ * V_; i += 256) xs[i] = xm[(size_t)n * C_ * V_ + i];
  __syncthreads();

  if (tid < E_ * V_) {
    const int e = tid / V_, v = tid - e * V_;
    float acc = 0.f;
    for (int c = 0; c < C_; ++c) acc += dyn_w[e * C_ + c] * xs[c * V_ + v];
    acc += dyn_b[e];
    const float sc = dyn_gamma[e] * rsqrtf(dyn_var[e] + EPS);
    acc = (acc - dyn_mean[e]) * sc + dyn_beta[e];
    Hs[e][v] = fmaxf(acc, 0.f);
  }
  __syncthreads();

  if (tid < E_) {                  // softmax over V per hyperedge (TEMPERATURE == 1)
    float mx = -1e30f;
    for (int v = 0; v < V_; ++v) mx = fmaxf(mx, Hs[tid][v]);
    float sum = 0.f;
    float ex[V_];
    for (int v = 0; v < V_; ++v) { ex[v] = __expf(Hs[tid][v] - mx); sum += ex[v]; }
    const float inv = 1.f / sum;
    for (int v = 0; v < V_; ++v) Hs[tid][v] = ex[v] * inv;
  } else if (tid >= 32 && tid < 32 + 5 * V_) {   // anatomical finger masks, rows 8..12
    const int idx = tid - 32, f = idx / V_, v = idx - f * V_;
    Hs[E_ + f][v] = (v >= 1 + 4 * f && v < 5 + 4 * f) ? 1.f : 0.f;
  }
  __syncthreads();

  for (int idx = tid; idx < VPAD * PCOLS; idx += 256) {
    const int r = idx / PCOLS, c2 = idx - r * PCOLS;
    float p = 0.f;
    if (r < V_ && c2 < V_) {
      for (int m = 0; m < M_; ++m) p += Hs[m][r] * Hs[m][c2];
    }
    Ppad[(size_t)n * VPAD * PCOLS + idx] = p;   // zero padding built in
  }
}

// ---------------------------------------------------------------- kernel 3
// Block = (tBlk in 0..15, oHalf in 0..1, n in 0..127), 256 threads = 8 waves.
// Stage 1: xf = relu(BN(ft_w @ x_tile))  via V_WMMA_F32_16X16X4_F32, K=128 in 32 steps.
// Stage 2: out = xf @ P + x_tile         via V_WMMA_F32_16X16X4_F32, K=VPAD in 6 steps.
__global__ __launch_bounds__(256) void kmain(
    const float* __restrict__ x, const float* __restrict__ ft_w,
    const float* __restrict__ ft_b, const float* __restrict__ ft_gamma,
    const float* __restrict__ ft_beta, const float* __restrict__ ft_mean,
    const float* __restrict__ ft_var, const float* __restrict__ Ppad,
    float* __restrict__ out) {
  __shared__ float xs[C_ * NCOL];          // 128*336 fp32 = 172 KB: x[n,:,t-tile,:]
  __shared__ float xfL[64 * TT * VPAD];    // 64*16*24 fp32 = 96 KB: xf, V zero-padded
  __shared__ float Pl[VPAD * PCOLS];       // 3 KB
  const int tBlk  = blockIdx.x;
  const int oHalf = blockIdx.y;
  const int n     = blockIdx.z;
  const int tid   = threadIdx.x;
  const int lane  = tid & 31, wave = tid >> 5;
  const int t0    = tBlk * TT;
  const int oBase = oHalf * 64;
  const int ln    = lane & 15;
  const int hi    = lane >> 4;     // lane half selects K pair (A/B) / row+8 (C/D)
  const int kb    = hi * 2;

  // ---- stage x tile + P via async direct-to-LDS (ASYNCcnt path); zero xfL meanwhile
  {
    const float* xsrc = x + ((size_t)(n * C_) * T_ + t0) * V_;
    for (int idx = tid; idx < C_ * NCOL; idx += 256) {     // rows are 84B-phased -> b32
      const int c = idx / NCOL, e = idx - c * NCOL;
      async_g2l_b32(xsrc + (size_t)c * T_ * V_ + e, &xs[idx]);
    }
    const float* psrc = Ppad + (size_t)n * VPAD * PCOLS;
    for (int idx = tid; idx < VPAD * PCOLS; idx += 256)
      async_g2l_b32(psrc + idx, &Pl[idx]);
    for (int idx = tid; idx < 64 * TT * VPAD; idx += 256) xfL[idx] = 0.f;
  }

  // ---- per-wave stage-1 setup: wave owns mTile = wave>>1 (16 o's), colHalf = wave&1
  const int mTile   = wave >> 1;
  const int colHalf = wave & 1;

  v2f afrag[32];   // A = ft_w rows for this o-tile, cached in regs (K=128 -> 32 groups of 4)
  {
    const int o = oBase + mTile * 16 + ln;
    const float* wp = ft_w + (size_t)o * C_;
#pragma unroll
    for (int kg = 0; kg < 32; ++kg) {
      afrag[kg].x = wp[kg * 4 + kb];
      afrag[kg].y = wp[kg * 4 + kb + 1];
    }
  }
  float fsv[8], fbv[8];   // fused BN constants per accumulator slot
#pragma unroll
  for (int i = 0; i < 8; ++i) {
    const int r = hi ? i + 8 : i;
    const int o = oBase + mTile * 16 + r;
    const float fs = ft_gamma[o] * rsqrtf(ft_var[o] + EPS);
    fsv[i] = fs;
    fbv[i] = (ft_b[o] - ft_mean[o]) * fs + ft_beta[o];
  }
  asm volatile("s_wait_asynccnt 0x0" ::: "memory");   // this wave's async copies landed
  __syncthreads();                                    // all waves' copies visible

  // ---- stage 1: 21 column tiles split 11/10 between the two waves of each mTile
  const int ct0 = colHalf ? 11 : 0;
  const int ct1 = colHalf ? 21 : 11;
  for (int ct = ct0; ct < ct1; ++ct) {
    v8f acc = {};
    const int colL = ct * 16 + ln;
#pragma unroll
    for (int kg = 0; kg < 32; ++kg) {
      v2f b;
      const int c0 = kg * 4 + kb;
      b.x = xs[c0 * NCOL + colL];
      b.y = xs[(c0 + 1) * NCOL + colL];
      acc = __builtin_amdgcn_wmma_f32_16x16x4_f32(false, afrag[kg], false, b,
                                                  (short)0, acc, false, false);
    }
    const int tl = colL / V_;
    const int v  = colL - tl * V_;
#pragma unroll
    for (int i = 0; i < 8; ++i) {
      const int r = hi ? i + 8 : i;
      const float val = fmaxf(acc[i] * fsv[i] + fbv[i], 0.f);   // BN + ReLU
      xfL[(mTile * 16 + r) * (TT * VPAD) + tl * VPAD + v] = val;
    }
  }
  __syncthreads();

  // ---- stage 2: out[o,t,v] = sum_v' xf[o,t,v'] * P[v',v] + x[o,t,v]
  // rows = (oLoc,tl): row-tile oLoc spans tl=0..15; wave owns oLoc = wave*8 .. wave*8+7
  for (int oo = 0; oo < 8; ++oo) {
    const int oLoc = wave * 8 + oo;
    v2f a2[6];
#pragma unroll
    for (int kg = 0; kg < 6; ++kg) {
      const int kk = kg * 4 + kb;
      a2[kg].x = xfL[oLoc * (TT * VPAD) + ln * VPAD + kk];
      a2[kg].y = xfL[oLoc * (TT * VPAD) + ln * VPAD + kk + 1];
    }
    const int oG = oBase + oLoc;
#pragma unroll
    for (int nt = 0; nt < 2; ++nt) {
      v8f acc = {};
#pragma unroll
      for (int kg = 0; kg < 6; ++kg) {
        v2f b;
        const int kk = kg * 4 + kb;
        b.x = Pl[kk * PCOLS + nt * 16 + ln];
        b.y = Pl[(kk + 1) * PCOLS + nt * 16 + ln];
        acc = __builtin_amdgcn_wmma_f32_16x16x4_f32(false, a2[kg], false, b,
                                                    (short)0, acc, false, false);
      }
      const int v = nt * 16 + ln;
      if (v < V_) {
#pragma unroll
        for (int i = 0; i < 8; ++i) {
          const int tl = hi ? i + 8 : i;
          const float res = xs[oG * NCOL + tl * V_ + v];   // residual from resident x tile
          out[(((size_t)n * C_ + oG) * T_ + (t0 + tl)) * V_ + v] = acc[i] + res;
        }
      }
    }
  }
}

// ---------------------------------------------------------------- launcher
extern "C" void kernel_launch(void* const* d_in, const int* in_sizes, int n_in,
                              void* d_out, int out_size, void* d_ws, size_t ws_size,
                              hipStream_t stream) {
  (void)in_sizes; (void)n_in; (void)out_size; (void)ws_size;
  const float* x         = (const float*)d_in[0];
  const float* dyn_w     = (const float*)d_in[1];
  const float* dyn_b     = (const float*)d_in[2];
  const float* dyn_gamma = (const float*)d_in[3];
  const float* dyn_beta  = (const float*)d_in[4];
  const float* dyn_mean  = (const float*)d_in[5];
  const float* dyn_var   = (const float*)d_in[6];
  const float* ft_w      = (const float*)d_in[7];
  const float* ft_b      = (const float*)d_in[8];
  const float* ft_gamma  = (const float*)d_in[9];
  const float* ft_beta   = (const float*)d_in[10];
  const float* ft_mean   = (const float*)d_in[11];
  const float* ft_var    = (const float*)d_in[12];
  float* ws = (float*)d_ws;          // needs (N*C*V + N*24*32)*4 = ~1.7 MB
  float* xm = ws + WS_XM;
  float* Pp = ws + WS_P;
  float* o  = (float*)d_out;

  kmean <<<N_ * C_, 256, 0, stream>>>(x, xm);
  kedges<<<N_,      256, 0, stream>>>(xm, dyn_w, dyn_b, dyn_gamma, dyn_beta,
                                      dyn_mean, dyn_var, Pp);
  kmain <<<dim3(T_ / TT, 2, N_), 256, 0, stream>>>(x, ft_w, ft_b, ft_gamma,
                                                   ft_beta, ft_mean, ft_var, Pp, o);
}